// ComputeVecSimilarityLoss_12506944766667
// MI455X (gfx1250) — compile-verified
//
#include <hip/hip_runtime.h>
#include <math.h>

typedef __attribute__((ext_vector_type(2))) float v2f;
typedef __attribute__((ext_vector_type(4))) float v4f;
typedef __attribute__((ext_vector_type(8))) float v8f;

#define B_    8
#define N_    48
#define M_    (N_ * N_)            // 2304
#define TILES (M_ / 16)            // 144 (exact)
#define EPS2  2e-9f                // per-element 1e-9 summed over D=2

static __device__ __forceinline__ unsigned wave_ballot32(bool p) {
#if __has_builtin(__builtin_amdgcn_ballot_w32)
    return __builtin_amdgcn_ballot_w32(p);
#else
    return (unsigned)__ballot(p);
#endif
}

// ds_swizzle group-of-32 xor pattern: offset = {0,xor_mask[14:10],or=0,and=0x1f}
#define SWZ_XOR(mask) (((mask) << 10) | 0x1f)

static __device__ __forceinline__ float wave32_sum(float x) {
    int xi;
    xi = __builtin_amdgcn_ds_swizzle(__float_as_int(x), SWZ_XOR(16)); x += __int_as_float(xi);
    xi = __builtin_amdgcn_ds_swizzle(__float_as_int(x), SWZ_XOR(8));  x += __int_as_float(xi);
    xi = __builtin_amdgcn_ds_swizzle(__float_as_int(x), SWZ_XOR(4));  x += __int_as_float(xi);
    xi = __builtin_amdgcn_ds_swizzle(__float_as_int(x), SWZ_XOR(2));  x += __int_as_float(xi);
    xi = __builtin_amdgcn_ds_swizzle(__float_as_int(x), SWZ_XOR(1));  x += __int_as_float(xi);
    return x;
}

// ---------------------------------------------------------------------------
// Kernel 1: build scaled difference vectors V[b][m][2] and INVERSE norms
// inv_vabs[b][m] = 1/sqrt(vx^2+vy^2+2e-9); zero accumulators.
// ---------------------------------------------------------------------------
__global__ void __launch_bounds__(256)
prep_kernel(const float* __restrict__ gt,   // [B][N][2]
            const float* __restrict__ cs,   // [B][N][N]
            const float* __restrict__ thr,  // [1]
            float* __restrict__ V,          // [B][M][2]
            float* __restrict__ ivabs,      // [B][M]
            float* __restrict__ acc)        // acc[0]=sum (f32), acc[1]=count (u32)
{
    int idx = blockIdx.x * blockDim.x + threadIdx.x;
    if (idx == 0) {
        acc[0] = 0.0f;
        ((unsigned int*)acc)[1] = 0u;
    }
    if (idx >= B_ * M_) return;

    int b = idx / M_;
    int m = idx % M_;
    int i = m / N_;
    int j = m % N_;

    float s = cs[(b * N_ + i) * N_ + j];
    float t = thr[0];
    s = (s < t) ? 0.0f : s;

    float vx = (gt[(b * N_ + i) * 2 + 0] - gt[(b * N_ + j) * 2 + 0]) * s;
    float vy = (gt[(b * N_ + i) * 2 + 1] - gt[(b * N_ + j) * 2 + 1]) * s;

    V[idx * 2 + 0] = vx;
    V[idx * 2 + 1] = vy;
    ivabs[idx] = 1.0f / sqrtf(vx * vx + vy * vy + EPS2);   // norm > 0 (eps)
}

// ---------------------------------------------------------------------------
// Kernel 2: grid (tj=TILES, tiGroup=TILES/8, b=B). 8 waves/block; wave w
// handles tile (ti = blockIdx.y*8 + w, tj = blockIdx.x) of batch blockIdx.z.
// One V_WMMA_F32_16X16X4_F32 per wave (K=2 zero-padded to 4).
// Epilogue:
//   count: per element, ballot(dot!=0) -> scalar popcount (SALU pipe)
//   sum:   fma(|dot|, ivm*ivn, sum), then 5x ds_swizzle xor-butterfly
//   block: 8 waves -> LDS -> 2 atomics per block
//
// A 16x4 layout : VGPR0 lanes0-15 = K=0 (M=lane), lanes16-31 = K=2 (zero pad)
//                 VGPR1 lanes0-15 = K=1,          lanes16-31 = K=3 (zero pad)
// B 4x16 layout : VGPR0 lanes0-15 = row K=0 (N=lane), lanes16-31 = K=2 (pad)
//                 VGPR1 lanes0-15 = row K=1,          lanes16-31 = K=3 (pad)
// D 16x16 layout: VGPR r: lanes0-15 -> (M=r, N=lane); lanes16-31 -> (M=r+8, N=lane-16)
// ---------------------------------------------------------------------------
__global__ void __launch_bounds__(256)
pairwise_kernel(const float* __restrict__ V,      // [B][M][2]
                const float* __restrict__ ivabs,  // [B][M]
                float* __restrict__ acc)
{
    __shared__ float        ssum[8];
    __shared__ unsigned int scnt[8];

    const int lane = threadIdx.x & 31;
    const int wave = threadIdx.x >> 5;

    const int tj = blockIdx.x;
    const int ti = blockIdx.y * 8 + wave;
    const int b  = blockIdx.z;

    const int half = lane >> 4;   // 0: K=0/1 halves, 1: K=2/3 zero-pad halves
    const int l    = lane & 15;

    const float* Vb  = V     + (size_t)b * M_ * 2;
    const float* ivb = ivabs + (size_t)b * M_;

    // --- load A (rows ti*16..+15) and B (rows tj*16..+15), K padded 2->4 ---
    v2f a, bmat;
    if (half == 0) {
        const float* pa = Vb + (ti * 16 + l) * 2;
        const float* pb = Vb + (tj * 16 + l) * 2;
        a.x    = pa[0];   // A: M=l, K=0
        a.y    = pa[1];   // A: M=l, K=1
        bmat.x = pb[0];   // B: K=0, N=l
        bmat.y = pb[1];   // B: K=1, N=l
    } else {
        a.x = 0.0f; a.y = 0.0f;               // K=2,3 zero pad
        bmat.x = 0.0f; bmat.y = 0.0f;
    }

    v8f c = {};
    v8f d = __builtin_amdgcn_wmma_f32_16x16x4_f32(
        /*neg_a=*/false, a, /*neg_b=*/false, bmat,
        /*c_mod=*/(short)0, c, /*reuse_a=*/false, /*reuse_b=*/false);

    // --- hoisted inverse norms: 8 consecutive row norms + 1 column norm ---
    const float  ivn = ivb[tj * 16 + l];                   // column n = tj*16+l
    const v4f*   pim = (const v4f*)(ivb + ti * 16 + 8 * half);
    const v4f    im0 = pim[0];                             // rows r=0..3 (+8*half)
    const v4f    im1 = pim[1];                             // rows r=4..7 (+8*half)

    float        sum = 0.0f;
    unsigned int cnt = 0u;                                 // wave-uniform (scalar)
#pragma unroll
    for (int r = 0; r < 8; ++r) {
        float val = d[r];
        float ivm = (r < 4) ? im0[r] : im1[r - 4];
        cnt += __popc(wave_ballot32(val != 0.0f));         // SALU: s_bcnt1 of mask
        sum  = fmaf(fabsf(val), ivm * ivn, sum);           // 0 contribution when val==0
    }

    // --- wave32 float butterfly: 5x ds_swizzle (xor), no clamp math ---
    sum = wave32_sum(sum);

    if (lane == 0) {
        ssum[wave] = sum;
        scnt[wave] = cnt;
    }
    __syncthreads();

    // --- block reduction: 8 partials -> 2 atomics per block ---
    if (threadIdx.x == 0) {
        float        ts = 0.0f;
        unsigned int tc = 0u;
#pragma unroll
        for (int i = 0; i < 8; ++i) {
            ts += ssum[i];
            tc += scnt[i];
        }
        atomicAdd(acc, ts);
        atomicAdd(((unsigned int*)acc) + 1, tc);
    }
}

// ---------------------------------------------------------------------------
// Kernel 3: out = sum / idx_num
// ---------------------------------------------------------------------------
__global__ void finalize_kernel(const float* __restrict__ acc, float* __restrict__ out)
{
    if (threadIdx.x == 0 && blockIdx.x == 0) {
        out[0] = acc[0] / (float)(((const unsigned int*)acc)[1]);
    }
}

// ---------------------------------------------------------------------------
extern "C" void kernel_launch(void* const* d_in, const int* in_sizes, int n_in,
                              void* d_out, int out_size, void* d_ws, size_t ws_size,
                              hipStream_t stream)
{
    const float* gt  = (const float*)d_in[0];   // [8][48][2]
    const float* cs  = (const float*)d_in[1];   // [8][48][48]
    const float* thr = (const float*)d_in[2];   // [1]

    float* V     = (float*)d_ws;                // B*M*2 floats
    float* ivabs = V + B_ * M_ * 2;             // B*M floats
    float* acc   = ivabs + B_ * M_;             // 2 slots (f32 sum, u32 count)
    float* out   = (float*)d_out;

    const int prep_n = B_ * M_;                               // 18432
    prep_kernel<<<(prep_n + 255) / 256, 256, 0, stream>>>(gt, cs, thr, V, ivabs, acc);

    dim3 grid(TILES, TILES / 8, B_);                          // (144, 18, 8)
    pairwise_kernel<<<grid, 256, 0, stream>>>(V, ivabs, acc);

    finalize_kernel<<<1, 1, 0, stream>>>(acc, out);
}